// MultiScaleFusionLayer_31653908972344
// MI455X (gfx1250) — compile-verified
//
#include <hip/hip_runtime.h>

// Problem constants (from reference)
#define B_N   2048
#define L_N   100
#define D_N   64
#define ROWS  16     // rows per block-tile = 4 batches * 4 l-positions

typedef __attribute__((ext_vector_type(2))) float v2f;
typedef __attribute__((ext_vector_type(8))) float v8f;

__device__ __forceinline__ v8f wmma4(v2f a, v2f b, v8f c) {
  // V_WMMA_F32_16X16X4_F32 : D = A(16x4) * B(4x16) + C(16x16)
  return __builtin_amdgcn_wmma_f32_16x16x4_f32(false, a, false, b, (short)0, c,
                                               false, false);
}

__device__ __forceinline__ v8f vzero8() {
  v8f z;
#pragma unroll
  for (int j = 0; j < 8; ++j) z[j] = 0.f;
  return z;
}

// MoE tile for ONE 16-col N-tile `n`: for 4 experts,
//   D_e = A(16xK) * W_e(Kx64)[:, n*16:(n+1)*16] + b_e
//   out[r][colBase + n*16 + c] = sum_e sel[r][e] * D_e[r][c]   (raw linear gate)
// 4 independent accumulator chains (one per expert) share each A-fragment.
template <int KSTEPS, int LDA>
__device__ __forceinline__ void moe_accum_n(const float (&A)[ROWS][LDA],
                                            const float* __restrict__ W,
                                            const float* __restrict__ bias,
                                            const float (&sel)[ROWS][4],
                                            float (&out)[ROWS][256], int colBase,
                                            int n, int lane) {
  const int rA   = lane & 15;
  const int koff = (lane >> 4) << 1;   // 0 for lanes 0-15, 2 for lanes 16-31
  const int colD = lane & 15;
  const int hi8  = (lane >> 4) << 3;   // D-tile row offset for this lane half
  const int cW   = n * 16 + colD;      // column in the 64-wide weight matrix

  v8f acc[4];
#pragma unroll
  for (int e = 0; e < 4; ++e) acc[e] = vzero8();

  for (int kk = 0; kk < KSTEPS; ++kk) {
    int kb = kk * 4 + koff;
    v2f a = *(const v2f*)&A[rA][kb];   // 8B aligned (kb even, LDA*4 % 8 == 0)
#pragma unroll
    for (int e = 0; e < 4; ++e) {
      const float* We = W + e * (KSTEPS * 4) * 64;
      v2f bf;
      bf.x = We[kb * 64 + cW];
      bf.y = We[(kb + 1) * 64 + cW];
      acc[e] = wmma4(a, bf, acc[e]);
    }
  }

  v8f gAcc = vzero8();
#pragma unroll
  for (int e = 0; e < 4; ++e) {
    float be = bias[e * 64 + cW];
    float selv[8];
#pragma unroll
    for (int j = 0; j < 8; ++j) selv[j] = sel[j + hi8][e];
#pragma unroll
    for (int j = 0; j < 8; ++j) gAcc[j] += selv[j] * (acc[e][j] + be);
  }
#pragma unroll
  for (int j = 0; j < 8; ++j) out[j + hi8][colBase + cW] = gAcc[j];
}

// Plain GEMM for one N-tile + bias: out[:, n*16:] = A(16xK) * W(Kx64)[:, n-tile] + bias
// K-chain split into 4 round-robin partial accumulators to keep 4 chains in flight.
template <int KSTEPS, int LDA>
__device__ __forceinline__ void gemm_bias_n(const float (&A)[ROWS][LDA],
                                            const float* __restrict__ W,
                                            const float* __restrict__ bias,
                                            float (&out)[ROWS][64], int n,
                                            int lane) {
  const int rA   = lane & 15;
  const int koff = (lane >> 4) << 1;
  const int colD = lane & 15;
  const int hi8  = (lane >> 4) << 3;
  const int cW   = n * 16 + colD;

  v8f acc[4];
#pragma unroll
  for (int p = 0; p < 4; ++p) acc[p] = vzero8();

  static_assert(KSTEPS % 4 == 0, "KSTEPS must be a multiple of 4");
  for (int kk4 = 0; kk4 < KSTEPS / 4; ++kk4) {
#pragma unroll
    for (int p = 0; p < 4; ++p) {
      int kb = (kk4 * 4 + p) * 4 + koff;
      v2f a = *(const v2f*)&A[rA][kb];
      v2f bf;
      bf.x = W[kb * 64 + cW];
      bf.y = W[(kb + 1) * 64 + cW];
      acc[p] = wmma4(a, bf, acc[p]);
    }
  }
  float bv = bias[cW];
#pragma unroll
  for (int j = 0; j < 8; ++j)
    out[j + hi8][cW] = (acc[0][j] + acc[1][j]) + (acc[2][j] + acc[3][j]) + bv;
}

__global__ void __launch_bounds__(128)
fusion_compute_kernel(const float* __restrict__ id_f, const float* __restrict__ img_f,
                      const float* __restrict__ txt_f,
                      const float* __restrict__ Ws,  const float* __restrict__ bs,
                      const float* __restrict__ Wt1, const float* __restrict__ bt1,
                      const float* __restrict__ Wt2, const float* __restrict__ bt2,
                      const float* __restrict__ Wt3, const float* __restrict__ bt3,
                      const float* __restrict__ Wgs, const float* __restrict__ bgs,
                      const float* __restrict__ Wg1, const float* __restrict__ bg1,
                      const float* __restrict__ Wg2, const float* __restrict__ bg2,
                      const float* __restrict__ Wg3, const float* __restrict__ bg3,
                      const float* __restrict__ Watt, const float* __restrict__ batt,
                      float* __restrict__ task_out /* [B_N][64] */) {
  __shared__ float xT[ROWS][192];    // concat(id|img|txt) for 16 (b,l) rows
  __shared__ float comb[ROWS][256];  // [g1 | g2 | g3 | gate_share]
  __shared__ float inT[ROWS][64];    // task-expert input, later attention logits
  __shared__ float selS[ROWS][4];
  __shared__ float mxT[ROWS], rsT[ROWS];

  const int tid  = threadIdx.x;      // 0..127 (4 waves)
  const int lane = tid & 31;
  const int w    = tid >> 5;         // N-tile owned by this wave (0..3)
  const int b0   = blockIdx.x * 4;   // 4 batches per block

  // ---- 1. stage X tile (only l = 0..3 matter) ----
  for (int i = tid; i < ROWS * 192; i += 128) {
    int r = i / 192, c = i - r * 192;
    int b = b0 + (r >> 2), l = r & 3;
    int base = (b * L_N + l) * D_N;
    float v;
    if (c < 64)       v = id_f[base + c];
    else if (c < 128) v = img_f[base + c - 64];
    else              v = txt_f[base + c - 128];
    xT[r][c] = v;
  }
  __syncthreads();

  // ---- 2. shared gate logits: selS[r][e] = x . Wgs[:,e] + bgs[e] ----
  if (tid < 64) {
    int r = tid >> 2, e = tid & 3;
    float s = bgs[e];
    for (int k = 0; k < 192; ++k) s += xT[r][k] * Wgs[k * 4 + e];
    selS[r][e] = s;
  }
  __syncthreads();

  // ---- 3. shared MoE -> comb[:,192:256]; wave w owns N-tile w ----
  moe_accum_n<48, 192>(xT, Ws, bs, selS, comb, 192, w, lane);
  __syncthreads();

  // ---- 4. task MoEs: g_t -> comb[:, t*64 : t*64+64] ----
#pragma unroll
  for (int t = 0; t < 3; ++t) {
    const float* Wt_ = (t == 0) ? Wt1 : (t == 1) ? Wt2 : Wt3;
    const float* bt_ = (t == 0) ? bt1 : (t == 1) ? bt2 : bt3;
    const float* Wg_ = (t == 0) ? Wg1 : (t == 1) ? Wg2 : Wg3;
    const float* bg_ = (t == 0) ? bg1 : (t == 1) ? bg2 : bg3;

    // expert input = feat_t + gate_share
    for (int i = tid; i < ROWS * 64; i += 128) {
      int r = i >> 6, c = i & 63;
      inT[r][c] = xT[r][t * 64 + c] + comb[r][192 + c];
    }
    // gate input = raw feat_t
    if (tid < 64) {
      int r = tid >> 2, e = tid & 3;
      float s = bg_[e];
      for (int k = 0; k < 64; ++k) s += xT[r][t * 64 + k] * Wg_[k * 4 + e];
      selS[r][e] = s;
    }
    __syncthreads();
    moe_accum_n<16, 64>(inT, Wt_, bt_, selS, comb, t * 64, w, lane);
    __syncthreads();
  }

  // ---- 5. attention logits = comb @ Watt + batt (reuse inT as logits) ----
  gemm_bias_n<64, 256>(comb, Watt, batt, inT, w, lane);
  __syncthreads();

  // ---- 6. softmax stats per row (over 64 cols) ----
  if (tid < ROWS) {
    float mx = -1e30f;
    for (int c = 0; c < 64; ++c) mx = fmaxf(mx, inT[tid][c]);
    float sm = 0.f;
    for (int c = 0; c < 64; ++c) sm += __expf(inT[tid][c] - mx);
    mxT[tid] = mx;
    rsT[tid] = 1.f / sm;
  }
  __syncthreads();

  // ---- 7. task[b][d] = sum_l G_l[b,0,d] * att[b,l,d], l=0..3 ----
#pragma unroll
  for (int q = 0; q < 2; ++q) {
    int idx = tid + q * 128;        // 0..255
    int bb = idx >> 6, d = idx & 63;
    int r0 = bb * 4;                // row of l==0 for this batch
    float acc = 0.f;
#pragma unroll
    for (int l = 0; l < 4; ++l) {
      float att = __expf(inT[r0 + l][d] - mxT[r0 + l]) * rsT[r0 + l];
      acc += comb[r0][l * 64 + d] * att;   // l: 0->g1, 1->g2, 2->g3, 3->gate_share
    }
    task_out[(b0 + bb) * 64 + d] = acc;
  }
}

// Broadcast task[B,64] -> out[B,100,64], coalesced float4 stores (~52 MB, BW bound)
__global__ void __launch_bounds__(256)
fusion_bcast_kernel(const float* __restrict__ task, float* __restrict__ out) {
  int i4 = blockIdx.x * 256 + threadIdx.x;       // one float4 each
  int base = i4 * 4;                              // < 13,107,200
  int b = base / (L_N * D_N);
  int rem = base - b * (L_N * D_N);
  int d = rem & 63;                               // multiple of 4
  float4 v = *(const float4*)(task + b * D_N + d);
  *(float4*)(out + base) = v;
}

extern "C" void kernel_launch(void* const* d_in, const int* in_sizes, int n_in,
                              void* d_out, int out_size, void* d_ws, size_t ws_size,
                              hipStream_t stream) {
  const float* id_f = (const float*)d_in[0];
  const float* img_f = (const float*)d_in[1];
  const float* txt_f = (const float*)d_in[2];
  const float* Ws  = (const float*)d_in[3];
  const float* bs  = (const float*)d_in[4];
  const float* Wt1 = (const float*)d_in[5];
  const float* bt1 = (const float*)d_in[6];
  const float* Wt2 = (const float*)d_in[7];
  const float* bt2 = (const float*)d_in[8];
  const float* Wt3 = (const float*)d_in[9];
  const float* bt3 = (const float*)d_in[10];
  const float* Wgs = (const float*)d_in[11];
  const float* bgs = (const float*)d_in[12];
  const float* Wg1 = (const float*)d_in[13];
  const float* bg1 = (const float*)d_in[14];
  const float* Wg2 = (const float*)d_in[15];
  const float* bg2 = (const float*)d_in[16];
  const float* Wg3 = (const float*)d_in[17];
  const float* bg3 = (const float*)d_in[18];
  const float* Watt = (const float*)d_in[19];
  const float* batt = (const float*)d_in[20];

  float* task_ws = (float*)d_ws;                 // B_N * 64 floats = 512 KB
  float* out = (float*)d_out;

  fusion_compute_kernel<<<B_N / 4, 128, 0, stream>>>(
      id_f, img_f, txt_f, Ws, bs, Wt1, bt1, Wt2, bt2, Wt3, bt3,
      Wgs, bgs, Wg1, bg1, Wg2, bg2, Wg3, bg3, Watt, batt, task_ws);

  int total4 = B_N * L_N * D_N / 4;              // 3,276,800
  fusion_bcast_kernel<<<total4 / 256, 256, 0, stream>>>(task_ws, out);
}